// YoloLossLayer_81423989997755
// MI455X (gfx1250) — compile-verified
//
#include <hip/hip_runtime.h>
#include <math.h>

#define IMGF   416.0f
#define NCLS   80
#define BATCH  32
#define KMAX   32
#define THRF   0.5f

typedef __attribute__((ext_vector_type(2))) float v2f;
typedef __attribute__((ext_vector_type(8))) float v8f;

// ---------------- workspace layout (float/u32 units) ----------------
// [0,   256)            acc[64][4]            (float)
// [256, 352)            counts[3][32]         (int)
// [352, 352+12288)      boxes[3][32][32][4]   (float)
// [12640, ...)          objectness bitmaps    (u32), per-scale bases

__device__ __forceinline__ float sigmoidf_(float x) {
    return 1.0f / (1.0f + __expf(-x));
}

__device__ __forceinline__ float bcef_(float l, float t) {
    return fmaxf(l, 0.0f) - l * t + log1pf(__expf(-fabsf(l)));
}

__device__ __forceinline__ float cliplogf_(float v) {
    v = fminf(fmaxf(v, 1e-9f), 1e9f);
    return __logf(v);
}

__device__ __forceinline__ float wave_reduce_(float v) {
    v += __shfl_down(v, 16, 32);
    v += __shfl_down(v, 8, 32);
    v += __shfl_down(v, 4, 32);
    v += __shfl_down(v, 2, 32);
    v += __shfl_down(v, 1, 32);
    return v;
}

// ---------------- phase 0: zero workspace prefix ----------------
__global__ void yolo_init_kernel(unsigned* __restrict__ ws, int nwords) {
    for (int i = blockIdx.x * blockDim.x + threadIdx.x; i < nwords;
         i += gridDim.x * blockDim.x)
        ws[i] = 0u;
}

// ---------------- phase 1: collect positive boxes + obj bitmap ----------------
__global__ void yolo_collect_kernel(const float* __restrict__ yt,
                                    int* __restrict__ count,      // [BATCH]
                                    float* __restrict__ boxes,    // [BATCH][KMAX][4]
                                    unsigned* __restrict__ bitmap,
                                    int G) {
    const int per_b = 3 * G * G;
    const int total = BATCH * per_b;
    int t = blockIdx.x * blockDim.x + threadIdx.x;
    if (t >= total) return;
    int b = t / per_b;
    const float* typ = yt + (size_t)t * 85;
    float obj = typ[4];
    if (obj > 0.0f) {
        atomicOr(&bitmap[t >> 5], 1u << (t & 31));
        int pos = atomicAdd(&count[b], 1);
        if (pos < KMAX) {
            float* dst = boxes + ((size_t)b * KMAX + pos) * 4;
            dst[0] = typ[0];
            dst[1] = typ[1];
            dst[2] = typ[2];
            dst[3] = typ[3];
        }
    }
}

// ---------------- phase 2: fused loss ----------------
__global__ void yolo_loss_kernel(const float* __restrict__ fm,
                                 const float* __restrict__ yt,
                                 const float* __restrict__ anch,   // 3x2 this scale
                                 const int* __restrict__ count,    // [BATCH]
                                 const float* __restrict__ boxes,  // [BATCH][KMAX][4]
                                 const unsigned* __restrict__ bitmap,
                                 float* __restrict__ acc,          // [64][4]
                                 int G, float stride) {
    const int GG = G * G;
    const int per_b = 3 * GG;
    const int total = BATCH * per_b;
    const int t = blockIdx.x * blockDim.x + threadIdx.x;

    float lxy = 0.0f, lwh = 0.0f, lconf = 0.0f, lcls = 0.0f;

    if (t < total) {
        const int b = t / per_b;
        const int r = t % per_b;
        const int a = r / GG;       // (b, a, cell) order: fm reads coalesced in cell
        const int cell = r % GG;
        const int gy = cell / G, gx = cell % G;

        const float* fp = fm + ((size_t)b * 255 + (size_t)a * 85) * GG + cell;
        const float p0 = fp[0];
        const float p1 = fp[(size_t)GG];
        const float p2 = fp[2 * (size_t)GG];
        const float p3 = fp[3 * (size_t)GG];
        const float p4 = fp[4 * (size_t)GG];

        const int bidx = b * per_b + cell * 3 + a;  // yt-order bit index
        const float om = ((bitmap[bidx >> 5] >> (bidx & 31)) & 1u) ? 1.0f : 0.0f;

        // predicted box (absolute coords)
        const float sx = sigmoidf_(p0), sy = sigmoidf_(p1);
        const float aw = anch[2 * a], ah = anch[2 * a + 1];
        const float bx = (sx + (float)gx) * stride;
        const float by = (sy + (float)gy) * stride;
        const float ew = __expf(p2), eh = __expf(p3);
        const float bw = ew * aw, bh = eh * ah;

        // ignore mask: best IoU vs this batch's positive boxes
        int cnt = count[b];
        if (cnt > KMAX) cnt = KMAX;
        const float* bb = boxes + (size_t)b * KMAX * 4;
        const float phw = 0.5f * bw, phh = 0.5f * bh;
        const float parea = bw * bh;
        float best = 0.0f;
        for (int i = 0; i < cnt; ++i) {
            const float tx = bb[i * 4 + 0], tyc = bb[i * 4 + 1];
            const float tw = bb[i * 4 + 2], th = bb[i * 4 + 3];
            const float thw = 0.5f * tw, thh = 0.5f * th;
            const float iw = fmaxf(fminf(bx + phw, tx + thw) - fmaxf(bx - phw, tx - thw), 0.0f);
            const float ih = fmaxf(fminf(by + phh, tyc + thh) - fmaxf(by - phh, tyc - thh), 0.0f);
            const float inter = iw * ih;
            const float iou = inter / (parea + tw * th - inter + 1e-10f);
            best = fmaxf(best, iou);
        }
        const float ign = (cnt > 0 && best < THRF) ? 1.0f : 0.0f;

        // confidence loss (all anchors)
        const float bce = bcef_(p4, om);
        const float df = om - sigmoidf_(p4);
        lconf = (om * bce + 0.5f * (1.0f - om) * ign * bce) * (df * df);

        // positive-only terms (object_mask multiplies them in the reference)
        if (om > 0.0f) {
            const float* typ = yt + ((size_t)b * per_b + (size_t)cell * 3 + a) * 85;
            const float ty0 = typ[0], ty1 = typ[1], ty2 = typ[2], ty3 = typ[3];
            const float inv = 1.0f / stride;
            const float bscale = 2.0f - (ty2 * (1.0f / IMGF)) * (ty3 * (1.0f / IMGF));

            const float dx = (ty0 * inv - (float)gx) - sx;
            const float dy = (ty1 * inv - (float)gy) - sy;
            lxy = 5.0f * bscale * (dx * dx + dy * dy);

            float ttw = ty2 / aw, tth = ty3 / ah;
            ttw = (ttw == 0.0f) ? 1.0f : ttw;
            tth = (tth == 0.0f) ? 1.0f : tth;
            const float ptw = (ew == 0.0f) ? 1.0f : ew;
            const float pth = (eh == 0.0f) ? 1.0f : eh;
            const float dw = cliplogf_(ttw) - cliplogf_(ptw);
            const float dh = cliplogf_(tth) - cliplogf_(pth);
            lwh = 5.0f * bscale * (dw * dw + dh * dh);

            const float* cp = fp + 5 * (size_t)GG;
            __builtin_prefetch(cp, 0, 0);
            float s = 0.0f;
            for (int c = 0; c < NCLS; ++c) {
                const float lg = cp[(size_t)c * GG];
                const float tt = 0.99f * typ[5 + c] + (0.01f / NCLS);
                s += bcef_(lg, tt);
            }
            lcls = s;
        }
    }

    // block reduction: wave shfl -> LDS -> one atomicAdd set per block
    __shared__ float red[8][4];
    float v[4] = {lxy, lwh, lconf, lcls};
    const int lane = threadIdx.x & 31;
    const int wv = threadIdx.x >> 5;
#pragma unroll
    for (int c = 0; c < 4; ++c) v[c] = wave_reduce_(v[c]);
    if (lane == 0) {
        red[wv][0] = v[0]; red[wv][1] = v[1];
        red[wv][2] = v[2]; red[wv][3] = v[3];
    }
    __syncthreads();
    if (threadIdx.x == 0) {
        float s[4] = {0.f, 0.f, 0.f, 0.f};
        const int nw = blockDim.x >> 5;
        for (int w = 0; w < nw; ++w)
            for (int c = 0; c < 4; ++c) s[c] += red[w][c];
        const int slot = (int)(blockIdx.x & 63u);
        for (int c = 0; c < 4; ++c) atomicAdd(&acc[slot * 4 + c], s[c]);
    }
}

// ---------------- phase 3: exact WMMA f32 reduction of 64 partials ----------------
// Step 1: A = 64 partials (16x4), B = ones(4x16)  -> D1 rows hold row-sums.
// Step 2: 4 accumulating WMMAs with A = ones(16x4), B = consecutive D1 VGPR
// pairs. Each (VGPR, lane-half) slot of D1 is consumed exactly once across the
// four calls, so every one of the 16 row-sums enters the accumulator exactly
// once -> exact f32 total, independent of the B-operand slot permutation.
__global__ void yolo_finalize_kernel(const float* __restrict__ acc,
                                     float* __restrict__ out) {
    const int lane = threadIdx.x;  // launched with exactly 32 threads (EXEC all 1s)
    v2f ones2;
    ones2.x = 1.0f;
    ones2.y = 1.0f;

    float totals[4];
#pragma unroll
    for (int c = 0; c < 4; ++c) {
        v2f amat;
        amat.x = acc[(2 * lane + 0) * 4 + c];
        amat.y = acc[(2 * lane + 1) * 4 + c];
        v8f zero = {};
        v8f d1 = __builtin_amdgcn_wmma_f32_16x16x4_f32(
            false, amat, false, ones2, (short)0, zero, false, false);
        v8f d2 = {};
#pragma unroll
        for (int i = 0; i < 4; ++i) {
            v2f bpair;
            bpair.x = d1[2 * i + 0];
            bpair.y = d1[2 * i + 1];
            d2 = __builtin_amdgcn_wmma_f32_16x16x4_f32(
                false, ones2, false, bpair, (short)0, d2, false, false);
        }
        totals[c] = d2[0];
    }

    if (lane == 0) {
        const float inv = 1.0f / (float)BATCH;
        const float xy = totals[0] * inv;
        const float wh = totals[1] * inv;
        const float cf = totals[2] * inv;
        const float pr = totals[3] * inv;
        out[0] = xy + wh + cf + pr;
        out[1] = xy;
        out[2] = wh;
        out[3] = cf;
        out[4] = pr;
    }
}

extern "C" void kernel_launch(void* const* d_in, const int* in_sizes, int n_in,
                              void* d_out, int out_size, void* d_ws, size_t ws_size,
                              hipStream_t stream) {
    (void)in_sizes; (void)n_in; (void)out_size; (void)ws_size;
    const float* fm[3] = {(const float*)d_in[0], (const float*)d_in[1], (const float*)d_in[2]};
    const float* yt[3] = {(const float*)d_in[3], (const float*)d_in[4], (const float*)d_in[5]};
    const float* anchors = (const float*)d_in[6];

    float* ws = (float*)d_ws;
    float* acc = ws;                                   // 256 floats
    int* counts = (int*)(ws + 256);                    // 96 ints
    float* boxes = ws + 352;                           // 3*32*32*4 floats
    unsigned* bitmap = (unsigned*)(ws + 352 + 12288);  // per-scale obj bitmaps

    const int Gs[3] = {13, 26, 52};
    int bmoff[3];
    int bmw = 0;
    for (int s = 0; s < 3; ++s) {
        bmoff[s] = bmw;
        bmw += (BATCH * 3 * Gs[s] * Gs[s] + 31) / 32;
    }

    // zero acc + counts + boxes + bitmaps
    const int zero_words = 256 + 96 + 12288 + bmw;
    yolo_init_kernel<<<(zero_words + 255) / 256, 256, 0, stream>>>(
        (unsigned*)d_ws, zero_words);

    for (int s = 0; s < 3; ++s) {
        const int G = Gs[s];
        const int total = BATCH * 3 * G * G;
        yolo_collect_kernel<<<(total + 255) / 256, 256, 0, stream>>>(
            yt[s], counts + s * BATCH, boxes + (size_t)s * BATCH * KMAX * 4,
            bitmap + bmoff[s], G);
    }

    for (int s = 0; s < 3; ++s) {
        const int G = Gs[s];
        const int total = BATCH * 3 * G * G;
        const float stride = IMGF / (float)G;
        const float* anch = anchors + (2 - s) * 6;  // s=0 -> [6:9], s=1 -> [3:6], s=2 -> [0:3]
        yolo_loss_kernel<<<(total + 255) / 256, 256, 0, stream>>>(
            fm[s], yt[s], anch, counts + s * BATCH,
            boxes + (size_t)s * BATCH * KMAX * 4, bitmap + bmoff[s], acc, G, stride);
    }

    yolo_finalize_kernel<<<1, 32, 0, stream>>>(acc, (float*)d_out);
}